// LambdaRankLoss_40355512713957
// MI455X (gfx1250) — compile-verified
//
#include <hip/hip_runtime.h>

#define TPB    256
#define ROW_N  1024
#define EPT    4      // ROW_N / TPB elements per thread
#define SPLIT  2      // sub-blocks per row
#define NWAVE  (TPB / 32)

// ---- CDNA5 async global->LDS staging (ASYNCcnt path), guarded ----
#if __has_builtin(__builtin_amdgcn_global_load_async_to_lds_b128)
#define HAVE_ASYNC_LDS 1
#endif

typedef int v4i __attribute__((ext_vector_type(4)));
typedef __attribute__((address_space(1))) v4i g_v4i;   // global
typedef __attribute__((address_space(3))) v4i l_v4i;   // LDS

__device__ __forceinline__ void stage_b128(const void* gsrc, void* ldst) {
#if defined(HAVE_ASYNC_LDS)
  __builtin_amdgcn_global_load_async_to_lds_b128((g_v4i*)gsrc, (l_v4i*)ldst, 0, 0);
#else
  *(uint4*)ldst = *(const uint4*)gsrc;
#endif
}

__device__ __forceinline__ void stage_wait() {
#if defined(HAVE_ASYNC_LDS)
#if __has_builtin(__builtin_amdgcn_s_wait_asynccnt)
  __builtin_amdgcn_s_wait_asynccnt(0);
#else
  asm volatile("s_wait_asynccnt 0" ::: "memory");
#endif
#endif
}

// Per-(row,sub) partial pair loss.
// Identity: sum_{i in P, j in N} softplus(s_j - s_i)
//         = ln2 * sum_{i,j} log2(e^{s_i} + e^{s_j})  -  Nneg * sum_{i in P} s_i
// 4 pairs share one v_log_f32 via log2(f0*f1*f2*f3); factors bounded well
// inside fp32 range for N(0,1) scores.
__global__ __launch_bounds__(TPB) void lambdarank_pairs(
    const float* __restrict__ scores, const int* __restrict__ rel,
    float* __restrict__ wsLoss, int* __restrict__ wsPairs) {
  __shared__ uint4 stageS4[ROW_N / 4];
  __shared__ uint4 stageR4[ROW_N / 4];
  __shared__ alignas(16) float ePos[ROW_N];
  __shared__ alignas(16) float eNeg[ROW_N];
  __shared__ int   waveTot[NWAVE];
  __shared__ float waveSum[NWAVE];
  __shared__ float waveAcc[NWAVE];

  const int tid  = threadIdx.x;
  const int lane = tid & 31;
  const int wid  = tid >> 5;
  const int row  = blockIdx.x / SPLIT;
  const int sub  = blockIdx.x % SPLIT;
  const size_t base = (size_t)row * ROW_N + (size_t)tid * EPT;

  __builtin_prefetch(scores + base, 0, 1);   // global_prefetch_b8
  __builtin_prefetch(rel + base, 0, 1);

  // Stage this row into LDS via async global->LDS copy.
  stage_b128(scores + base, &stageS4[tid]);
  stage_b128(rel + base, &stageR4[tid]);
  stage_wait();
  __syncthreads();                                        // barrier #1

  const float4 s4 = ((const float4*)stageS4)[tid];
  const uint4  r4 = stageR4[tid];
  float    sv[EPT] = {s4.x, s4.y, s4.z, s4.w};
  unsigned rv[EPT] = {r4.x, r4.y, r4.z, r4.w};

  // ---- deterministic compaction: wave32 shuffle scan + 8-entry combine ----
  const int cnt = (rv[0] == 1u) + (rv[1] == 1u) + (rv[2] == 1u) + (rv[3] == 1u);
  int x = cnt;
  #pragma unroll
  for (int d = 1; d < 32; d <<= 1) {
    int y = __shfl_up(x, d, 32);
    if (lane >= d) x += y;                 // inclusive scan within wave
  }
  if (lane == 31) waveTot[wid] = x;
  __syncthreads();                                        // barrier #2
  int wOff = 0, Ptot = 0;
  #pragma unroll
  for (int w = 0; w < NWAVE; ++w) {
    const int t = waveTot[w];
    Ptot += t;
    if (w < wid) wOff += t;
  }
  const int P  = Ptot;
  const int Ng = ROW_N - P;                // rel is {0,1}: neg = complement
  int pi = wOff + x - cnt;                 // exclusive prefix, index-ordered
  int ni = tid * EPT - pi;

  float sumPos = 0.0f;
  #pragma unroll
  for (int q = 0; q < EPT; ++q) {
    const float e = __expf(sv[q]);         // one exp per element, reused by all pairs
    if (rv[q] == 1u) { ePos[pi++] = e; sumPos += sv[q]; }
    else             { eNeg[ni++] = e; }
  }

  // wave-level reduce of sumPos (fixed butterfly order -> deterministic)
  float sp = sumPos;
  #pragma unroll
  for (int m = 16; m > 0; m >>= 1) sp += __shfl_xor(sp, m, 32);
  if (lane == 0) waveSum[wid] = sp;
  __syncthreads();   // barrier #3: ePos/eNeg/waveSum visible to all waves

  // ---- pair loop: this block owns negatives n == sub (mod SPLIT) ----
  const int P16 = P & ~15;
  float acc = 0.0f;
  for (int n = sub + SPLIT * tid; n < Ng; n += SPLIT * TPB) {
    const float en = eNeg[n];
    float a0 = 0.f, a1 = 0.f, a2 = 0.f, a3 = 0.f;
    int p = 0;
    for (; p < P16; p += 16) {             // 16 pairs: 4x ds_load_b128, 4 logs
      const float4 x0 = *(const float4*)&ePos[p];
      const float4 x1 = *(const float4*)&ePos[p + 4];
      const float4 x2 = *(const float4*)&ePos[p + 8];
      const float4 x3 = *(const float4*)&ePos[p + 12];
      a0 += __log2f(((x0.x + en) * (x0.y + en)) * ((x0.z + en) * (x0.w + en)));
      a1 += __log2f(((x1.x + en) * (x1.y + en)) * ((x1.z + en) * (x1.w + en)));
      a2 += __log2f(((x2.x + en) * (x2.y + en)) * ((x2.z + en) * (x2.w + en)));
      a3 += __log2f(((x3.x + en) * (x3.y + en)) * ((x3.z + en) * (x3.w + en)));
    }
    for (; p < P; ++p) a0 += __log2f(ePos[p] + en);
    acc += (a0 + a1) + (a2 + a3);
  }

  // wave-level reduce of acc + 8-entry combine
  float av = acc;
  #pragma unroll
  for (int m = 16; m > 0; m >>= 1) av += __shfl_xor(av, m, 32);
  if (lane == 0) waveAcc[wid] = av;
  __syncthreads();                                        // barrier #4
  if (tid == 0) {
    float tot = 0.0f;
    #pragma unroll
    for (int w = 0; w < NWAVE; ++w) tot += waveAcc[w];
    float partial = tot * 0.69314718055994530942f;        // ln2 * sum(log2)
    if (sub == 0) {
      float spAll = 0.0f;
      #pragma unroll
      for (int w = 0; w < NWAVE; ++w) spAll += waveSum[w];
      partial -= (float)Ng * spAll;                       // the "- s_i" term, once per row
      wsPairs[row] = P * Ng;
    }
    wsLoss[blockIdx.x] = partial;
  }
}

__global__ __launch_bounds__(TPB) void lambdarank_finalize(
    const float* __restrict__ wsLoss, const int* __restrict__ wsPairs,
    float* __restrict__ out, int nLoss, int nPairs) {
  __shared__ float sl[TPB];
  __shared__ int   sp[TPB];
  const int t = threadIdx.x;
  float a = 0.0f;
  for (int i = t; i < nLoss; i += TPB) a += wsLoss[i];
  int pc = 0;
  for (int i = t; i < nPairs; i += TPB) pc += wsPairs[i];
  sl[t] = a; sp[t] = pc;
  __syncthreads();
  for (int st = TPB / 2; st > 0; st >>= 1) {
    if (t < st) { sl[t] += sl[t + st]; sp[t] += sp[t + st]; }
    __syncthreads();
  }
  if (t == 0) {
    const int np = sp[0];
    out[0] = (np > 0) ? (sl[0] / (float)np) : 0.0f;
  }
}

extern "C" void kernel_launch(void* const* d_in, const int* in_sizes, int n_in,
                              void* d_out, int out_size, void* d_ws, size_t ws_size,
                              hipStream_t stream) {
  const float* scores = (const float*)d_in[0];
  const int*   rel    = (const int*)d_in[1];
  const int rows = in_sizes[0] / ROW_N;            // 128

  float* wsLoss  = (float*)d_ws;                   // rows*SPLIT floats
  int*   wsPairs = (int*)((char*)d_ws + (size_t)rows * SPLIT * sizeof(float));

  lambdarank_pairs<<<rows * SPLIT, TPB, 0, stream>>>(scores, rel, wsLoss, wsPairs);
  lambdarank_finalize<<<1, TPB, 0, stream>>>(wsLoss, wsPairs, (float*)d_out,
                                             rows * SPLIT, rows);
}